// Basic_Model_19078244729512
// MI455X (gfx1250) — compile-verified
//
#include <hip/hip_runtime.h>

#define TILE 256

typedef int v4i __attribute__((vector_size(16)));
typedef __attribute__((address_space(1))) v4i* gas_v4i_ptr;
typedef __attribute__((address_space(3))) v4i* las_v4i_ptr;

#if defined(__gfx1250__) && __has_builtin(__builtin_amdgcn_global_load_async_to_lds_b128)
#define USE_ASYNC 1
#else
#define USE_ASYNC 0
#endif

__device__ __forceinline__ void wait_asynccnt0() {
#if defined(__gfx1250__)
#if __has_builtin(__builtin_amdgcn_s_wait_asynccnt)
    __builtin_amdgcn_s_wait_asynccnt(0);
#else
    asm volatile("s_wait_asynccnt 0" ::: "memory");
#endif
#endif
}

__global__ __launch_bounds__(TILE) void returning_rate_kernel(
    const float* __restrict__ out_in,
    const float* __restrict__ mask_in,
    float* __restrict__ p1_out,
    float* __restrict__ p2_out,
    int B)
{
    __shared__ __align__(16) float s_out [TILE * 25];
    __shared__ __align__(16) float s_mask[TILE * 25];
    __shared__ __align__(16) float s_p1  [TILE * 5];
    __shared__ __align__(16) float s_p2  [TILE * 5];

    const int tid  = threadIdx.x;
    const int base = blockIdx.x * TILE;           // B = 2e6 -> fits in int
    const int rem  = min(TILE, B - base);
    if (rem <= 0) return;

    const float* __restrict__ g0 = out_in  + (size_t)base * 25;
    const float* __restrict__ g1 = mask_in + (size_t)base * 25;

    // ---- Stage tile into LDS: coalesced 128-bit async granules ----
    const int nfl = rem * 25;
    const int n4  = nfl >> 2;                     // float4 count
    for (int i = tid; i < n4; i += TILE) {
#if USE_ASYNC
        __builtin_amdgcn_global_load_async_to_lds_b128(
            (gas_v4i_ptr)(uintptr_t)(g0 + (size_t)i * 4),
            (las_v4i_ptr)(&s_out[(size_t)i * 4]),
            0, 0);
        __builtin_amdgcn_global_load_async_to_lds_b128(
            (gas_v4i_ptr)(uintptr_t)(g1 + (size_t)i * 4),
            (las_v4i_ptr)(&s_mask[(size_t)i * 4]),
            0, 0);
#else
        ((float4*)s_out )[i] = ((const float4*)g0)[i];
        ((float4*)s_mask)[i] = ((const float4*)g1)[i];
#endif
    }
    // scalar tail (nfl may not be a multiple of 4 for a partial tile)
    for (int i = (n4 << 2) + tid; i < nfl; i += TILE) {
        s_out [i] = g0[i];
        s_mask[i] = g1[i];
    }
#if USE_ASYNC
    wait_asynccnt0();
#endif
    __syncthreads();

    // ---- Per-sample compute from LDS (pitch 25 is coprime with 64 banks) ----
    if (tid < rem) {
        const float* r = &s_out [tid * 25];
        const float* m = &s_mask[tid * 25];

        const float p40 = r[20], p41 = r[21], p42 = r[22], p43 = r[23], p44 = r[24];
        const float p31 = r[16], p32 = r[17], p33 = r[18], p34 = r[19];
        const float p22 = r[12], p23 = r[13], p24 = r[14];
        const float p13 = r[8],  p14 = r[9];
        const float p04 = r[4];
        const float m31 = m[16], m22 = m[12], m13 = m[8], m04 = m[4];

        // p1: mask multiplies only the last product term (reference precedence)
        const float p1_1 = p40 * p41 * m31;
        const float p1_2 = p40 * p42 + p31 * p32 * m22;
        const float p1_3 = p40 * p43 + p31 * p33 + p22 * p23 * m13;
        const float p1_4 = p40 * p44 + p31 * p34 + p22 * p24 + p13 * p14 * m04;

        // p2: complement chains
        const float q40 = 1.0f - p40;
        const float q31 = 1.0f - p31;
        const float q22 = 1.0f - p22;
        const float q13 = 1.0f - p13;
        const float p2_1 = q31 * (1.0f - q40) * m31;
        const float p2_2 = q22 * (1.0f - q40 * q31) * m22;
        const float p2_3 = q13 * (1.0f - q40 * q31 * q22) * m13;
        const float p2_4 = (1.0f - p04) * (1.0f - q40 * q31 * q22 * q13) * m04;

        float* o1 = &s_p1[tid * 5];
        o1[0] = 0.0f; o1[1] = p1_1; o1[2] = p1_2; o1[3] = p1_3; o1[4] = p1_4;
        float* o2 = &s_p2[tid * 5];
        o2[0] = 0.0f; o2[1] = p2_1; o2[2] = p2_2; o2[3] = p2_3; o2[4] = p2_4;
    }
    __syncthreads();

    // ---- Coalesced writeback ----
    float* __restrict__ d1 = p1_out + (size_t)base * 5;
    float* __restrict__ d2 = p2_out + (size_t)base * 5;
    const int nout = rem * 5;
    for (int i = tid; i < nout; i += TILE) {
        d1[i] = s_p1[i];
        d2[i] = s_p2[i];
    }
}

extern "C" void kernel_launch(void* const* d_in, const int* in_sizes, int n_in,
                              void* d_out, int out_size, void* d_ws, size_t ws_size,
                              hipStream_t stream) {
    (void)n_in; (void)d_ws; (void)ws_size;
    const float* out_in  = (const float*)d_in[0];
    const float* mask_in = (const float*)d_in[1];
    const int B = in_sizes[0] / 25;

    float* p1 = (float*)d_out;                 // p_1 flat [B,5]
    float* p2 = p1 + (size_t)B * 5;            // p_2 flat [B,5]
    (void)out_size;

    const int grid = (B + TILE - 1) / TILE;
    returning_rate_kernel<<<grid, TILE, 0, stream>>>(out_in, mask_in, p1, p2, B);
}